// ManifoldLagrangeLoss_55817394979270
// MI455X (gfx1250) — compile-verified
//
#include <hip/hip_runtime.h>

// ManifoldLagrangeLoss (Sn): loss = (1/(N*D)) * sum_rows (||y_row|| - 1)^2
// since sum_d (y/||y|| - y)^2 == (||y||-1)^2 per row. yinds (d_in[1]) is unused
// by the reference. Output: single f32 scalar.
//
// Memory-bound: 256 MB read @ 23.3 TB/s ~= 11 us floor. Row norms computed on
// the matrix pipe: G = Y*Y^T per 16x16 tile via 4x V_WMMA_F32_16X16X4_F32;
// the f32 A(16x4) and B(4x16) VGPR layouts are mutual transposes, so the same
// per-lane float2 feeds both operands. Diagonal of G = row norms^2.

typedef float v2f __attribute__((ext_vector_type(2)));
typedef float v8f __attribute__((ext_vector_type(8)));

#define NROWS   4000000
#define DFEAT   16
#define NTILES  (NROWS / 16)          // 250000 tiles of 16 rows (16x16 f32 = 1 KB)
#define BLOCKS  2048
#define TPB     256                   // 8 wave32 per block
#define WPB     (TPB / 32)
#define TOTAL_WAVES (BLOCKS * WPB)    // 16384 waves -> ~15.3 tiles/wave

__global__ __launch_bounds__(TPB) void mll_pass1(const float* __restrict__ y1,
                                                 float* __restrict__ partials) {
  const int lane = threadIdx.x & 31;
  const int wib  = threadIdx.x >> 5;                  // wave-in-block
  // Wave id is identical across the 32 lanes; readfirstlane pins it to an SGPR
  // so the tile loop becomes scalar control flow (s_cmp/s_cbranch, EXEC
  // untouched) -> WMMA's EXEC==all-ones requirement holds structurally.
  const int wave = __builtin_amdgcn_readfirstlane(blockIdx.x * WPB + wib);

  const int row  = lane & 15;                         // row within tile (A: M, B: N)
  const int half = lane >> 4;                         // K pair select (0 or 1)
  // Diagonal of 16x16 f32 D-matrix: lanes 0-7 hold G[m][m] in g[lane] (M=lane),
  // lanes 24-31 hold G[m][m] in g[lane-24] (M=lane-16). Other lanes: off-diag.
  const bool diagActive = (lane < 8) || (lane >= 24);
  const int  diagIdx    = lane & 7;

  float partial = 0.0f;

  for (int t = wave; t < NTILES; t += TOTAL_WAVES) {
    // Lane's slice of the tile: row `row`, columns {4c + 2*half, 4c + 2*half+1}.
    // This float2 is simultaneously the A(16x4) and B(4x16) register image
    // (B = A^T, and the ISA layouts are transposes of each other).
    const float* base = y1 + (size_t)t * (16 * DFEAT) + (size_t)row * DFEAT + 2 * half;

    // Warm GL2 for this wave's next tile (speculative; OOB silently dropped).
    __builtin_prefetch((const void*)(base + (size_t)TOTAL_WAVES * (16 * DFEAT)), 0, 1);

    // One-shot 256 MB stream: non-temporal so we don't churn L2 residency.
    v2f a0 = __builtin_nontemporal_load((const v2f*)(base + 0));   // K = 0..3
    v2f a1 = __builtin_nontemporal_load((const v2f*)(base + 4));   // K = 4..7
    v2f a2 = __builtin_nontemporal_load((const v2f*)(base + 8));   // K = 8..11
    v2f a3 = __builtin_nontemporal_load((const v2f*)(base + 12));  // K = 12..15

    v8f g = {0.f, 0.f, 0.f, 0.f, 0.f, 0.f, 0.f, 0.f};
    // G += Y_c * Y_c^T, c = 0..3  (accumulates full 16x16x16 Gram matrix)
    g = __builtin_amdgcn_wmma_f32_16x16x4_f32(false, a0, false, a0, (short)0, g, false, false);
    g = __builtin_amdgcn_wmma_f32_16x16x4_f32(false, a1, false, a1, (short)0, g, false, false);
    g = __builtin_amdgcn_wmma_f32_16x16x4_f32(false, a2, false, a2, (short)0, g, false, false);
    g = __builtin_amdgcn_wmma_f32_16x16x4_f32(false, a3, false, a3, (short)0, g, false, false);

    // Extract this lane's diagonal element (8 cndmasks; cmp masks hoisted).
    float s = 0.0f;
#pragma unroll
    for (int r = 0; r < 8; ++r) s = (diagIdx == r) ? g[r] : s;

    // Raw v_sqrt_f32: inputs are chi-square sums, well inside normal range,
    // so skip the precise-sqrt denorm fixup sequence.
    float e = __builtin_amdgcn_sqrtf(s) - 1.0f;
    partial += diagActive ? (e * e) : 0.0f;
  }

  // Deterministic wave32 reduction.
#pragma unroll
  for (int off = 16; off > 0; off >>= 1)
    partial += __shfl_xor(partial, off, 32);

  __shared__ float wred[WPB];
  if (lane == 0) wred[wib] = partial;
  __syncthreads();
  if (threadIdx.x == 0) {
    float s = 0.0f;
#pragma unroll
    for (int i = 0; i < WPB; ++i) s += wred[i];
    partials[blockIdx.x] = s;
  }
}

__global__ __launch_bounds__(TPB) void mll_pass2(const float* __restrict__ partials,
                                                 float* __restrict__ out, int n) {
  const int lane = threadIdx.x & 31;
  const int wib  = threadIdx.x >> 5;
  float s = 0.0f;
  for (int i = threadIdx.x; i < n; i += TPB)  // fixed order -> deterministic
    s += partials[i];
#pragma unroll
  for (int off = 16; off > 0; off >>= 1)
    s += __shfl_xor(s, off, 32);
  __shared__ float wred[WPB];
  if (lane == 0) wred[wib] = s;
  __syncthreads();
  if (threadIdx.x == 0) {
    float tot = 0.0f;
#pragma unroll
    for (int i = 0; i < WPB; ++i) tot += wred[i];
    // lambda1 = 1.0; mean over N*D elements
    out[0] = tot * (1.0f / ((float)NROWS * (float)DFEAT));
  }
}

extern "C" void kernel_launch(void* const* d_in, const int* in_sizes, int n_in,
                              void* d_out, int out_size, void* d_ws, size_t ws_size,
                              hipStream_t stream) {
  const float* y1 = (const float*)d_in[0];
  (void)in_sizes; (void)n_in; (void)out_size; (void)ws_size;  // yinds (d_in[1]) unused
  float* partials = (float*)d_ws;  // BLOCKS floats = 8 KB scratch

  mll_pass1<<<BLOCKS, TPB, 0, stream>>>(y1, partials);
  mll_pass2<<<1, TPB, 0, stream>>>(partials, (float*)d_out, BLOCKS);
}